// DynamicConv1dTBC_11081015624097
// MI455X (gfx1250) — compile-verified
//
#include <hip/hip_runtime.h>
#include <hip/hip_bf16.h>

// Problem constants (from reference)
#define TT   1024
#define BB   16
#define CC   1024
#define HH   16
#define KK   7
#define PADL 6
#define RR   (CC / HH)        // 64
#define OO   (HH * KK)        // 112
#define MM   (TT * BB)        // 16384

typedef __attribute__((ext_vector_type(2))) float v2f;
typedef __attribute__((ext_vector_type(4))) float v4f;
typedef __attribute__((ext_vector_type(8))) float v8f;

// ---------------------------------------------------------------------------
// Kernel 1: logits[m, o] = sum_c x[m, c] * W[o, c]   (M=16384, N=112, K=1024)
// One wave computes a 16(M) x 112(N) tile via 7 fp32 WMMA accumulators.
// 8 waves per block -> 128 rows of M per block -> grid.x = 128.
// ---------------------------------------------------------------------------
__global__ __launch_bounds__(256)
void gemm_logits_wmma(const float* __restrict__ x, const float* __restrict__ W,
                      float* __restrict__ logits) {
    const int lane  = threadIdx.x & 31;
    const int wave  = threadIdx.x >> 5;                 // 0..7
    const int m0    = (blockIdx.x * 8 + wave) * 16;     // 16-row M tile base
    const int mrow  = m0 + (lane & 15);                 // this lane's A row
    const int khalf = (lane >> 4) * 2;                  // K sub-offset {0,2}
    const int ncol  = lane & 15;                        // this lane's B/C column

    const float* __restrict__ aptr = x + (size_t)mrow * CC + khalf;
    const float* __restrict__ bptr[7];
#pragma unroll
    for (int j = 0; j < 7; ++j)
        bptr[j] = W + (size_t)(j * 16 + ncol) * CC + khalf;

    v8f acc[7] = {};

    // K loop: 1024 / 4 = 256 steps; 1 A b64 load + 7 B b64 loads + 7 WMMA per step
#pragma unroll 4
    for (int k0 = 0; k0 < CC; k0 += 4) {
        v2f a = *(const v2f*)(aptr + k0);
#pragma unroll
        for (int j = 0; j < 7; ++j) {
            v2f b = *(const v2f*)(bptr[j] + k0);
            acc[j] = __builtin_amdgcn_wmma_f32_16x16x4_f32(
                false, a, false, b, (short)0, acc[j], false, false);
        }
    }

    // C/D layout: lane%16 = N; VGPR v = M (lanes 16-31: M = v + 8)
    float* __restrict__ wrow = logits + (size_t)(m0 + (lane >> 4) * 8) * OO + ncol;
#pragma unroll
    for (int j = 0; j < 7; ++j) {
#pragma unroll
        for (int v = 0; v < 8; ++v)
            wrow[(size_t)v * OO + j * 16] = acc[j][v];
    }
}

// ---------------------------------------------------------------------------
// Kernel 2: masked softmax over K=7 per (m, h). Invalid (pos<0) -> 0.
// ---------------------------------------------------------------------------
__global__ __launch_bounds__(256)
void softmax7(const float* __restrict__ logits, float* __restrict__ probs) {
    const int idx = blockIdx.x * 256 + threadIdx.x;     // (m, h)
    if (idx >= MM * HH) return;
    const int m = idx >> 4;
    const int h = idx & 15;
    const int t = m / BB;                               // m = t*B + b
    const int kmin = (t >= PADL) ? 0 : (PADL - t);      // pos = t + k - PADL >= 0

    const float* __restrict__ lg = logits + (size_t)m * OO + h * KK;
    float v[KK];
    float mx = -__builtin_inff();
#pragma unroll
    for (int k = 0; k < KK; ++k) {
        v[k] = lg[k];
        if (k >= kmin) mx = fmaxf(mx, v[k]);
    }
    float e[KK];
    float s = 0.f;
#pragma unroll
    for (int k = 0; k < KK; ++k) {
        e[k] = (k >= kmin) ? expf(v[k] - mx) : 0.f;
        s += e[k];
    }
    const float inv = 1.f / s;
    float* __restrict__ pr = probs + (size_t)m * OO + h * KK;
#pragma unroll
    for (int k = 0; k < KK; ++k) pr[k] = e[k] * inv;
}

// ---------------------------------------------------------------------------
// Kernel 3: out[t,b,h,r] = sum_k probs[t,b,h,k] * x[t+k-PADL, b, h*R + r]
// One thread = one float4 of out. x viewed as float4 (4096 float4 per t).
// ---------------------------------------------------------------------------
__global__ __launch_bounds__(256)
void band_out(const v4f* __restrict__ x4, const float* __restrict__ probs,
              v4f* __restrict__ out4) {
    const int idx = blockIdx.x * 256 + threadIdx.x;
    if (idx >= TT * BB * (CC / 4)) return;
    const int t  = idx >> 12;            // / (BB * 256)
    const int r  = idx & 4095;
    const int b  = r >> 8;
    const int c4 = r & 255;              // float4 index within C
    const int h  = c4 >> 4;              // 16 float4 per head

    const float* __restrict__ p = probs + (size_t)(t * BB + b) * OO + h * KK;
    v4f acc = {};
#pragma unroll
    for (int k = 0; k < KK; ++k) {
        const int pos = t + k - PADL;
        if (pos >= 0) {
            const float pk = p[k];
            v4f xv = x4[(size_t)pos * (BB * 256) + b * 256 + c4];
            acc += pk * xv;
        }
    }
    out4[idx] = acc;
}

// ---------------------------------------------------------------------------
// Kernel 4: dense[bh, t, c] = probs at band columns, 0 elsewhere (1 GiB).
// One wave per row (1024 cols); 8 float4 NT stores per lane, coalesced.
// ---------------------------------------------------------------------------
__global__ __launch_bounds__(256)
void dense_fill(const float* __restrict__ probs, float* __restrict__ dense) {
    const int lane = threadIdx.x & 31;
    const int wave = threadIdx.x >> 5;
    const int row  = blockIdx.x * 8 + wave;             // row = bh*T + t
    const int t  = row & (TT - 1);
    const int bh = row >> 10;
    const int b  = bh >> 4;
    const int h  = bh & 15;

    const float* __restrict__ p = probs + (size_t)(t * BB + b) * OO + h * KK;
    float* __restrict__ drow = dense + (size_t)row * TT;

    const int lo = t - PADL;                            // band is cols [lo, t]
#pragma unroll
    for (int i = 0; i < 8; ++i) {
        const int g  = i * 32 + lane;                   // float4 group (coalesced)
        const int c0 = g * 4;
        v4f val = {};
        if (c0 + 3 >= lo && c0 <= t) {                  // rare: band overlap
#pragma unroll
            for (int j = 0; j < 4; ++j) {
                const int k = (c0 + j) - lo;            // k = c - t + PADL
                if (k >= 0 && k < KK) val[j] = p[k];
            }
        }
        __builtin_nontemporal_store(val, (v4f*)(drow + c0));
    }
}

// ---------------------------------------------------------------------------
extern "C" void kernel_launch(void* const* d_in, const int* in_sizes, int n_in,
                              void* d_out, int out_size, void* d_ws, size_t ws_size,
                              hipStream_t stream) {
    const float* x = (const float*)d_in[0];   // (T, B, C)
    const float* W = (const float*)d_in[1];   // (H*K, C)

    float* out   = (float*)d_out;                              // T*B*C floats
    float* dense = (float*)d_out + (size_t)TT * BB * CC;       // (B*H, T, T)

    float* logits = (float*)d_ws;                              // (M, 112)
    float* probs  = logits + (size_t)MM * OO;                  // (M, 112)

    // 1) WMMA GEMM: logits = x @ W^T
    gemm_logits_wmma<<<MM / 128, 256, 0, stream>>>(x, W, logits);

    // 2) masked softmax over K=7
    softmax7<<<(MM * HH) / 256, 256, 0, stream>>>(logits, probs);

    // 3) banded weighted sum -> out
    band_out<<<(TT * BB * (CC / 4)) / 256, 256, 0, stream>>>(
        (const v4f*)x, probs, (v4f*)out);

    // 4) dense band matrix (bandwidth-dominant, non-temporal streaming stores)
    dense_fill<<<(BB * HH * TT) / 8, 256, 0, stream>>>(probs, dense);
}